// MonotoneRQTransform_73538430042403
// MI455X (gfx1250) — compile-verified
//
#include <hip/hip_runtime.h>
#include <stdint.h>

#define TAILV 3.0f
#define EPSV 1e-6f
#define MIN_BW 0.01f
#define MIN_BH 0.01f
#define MIN_D 0.01f
#define CCH 128
#define KB 8

typedef float v4f __attribute__((ext_vector_type(4)));
typedef __attribute__((address_space(3))) float lds_float;

// ---------------------------------------------------------------------------
// Prep: per-channel softmax/cumsum/softplus -> packed tables in workspace.
//   edges_g  [C*K]    : cw[1..8] per channel (search keys)
//   params_g [C*K*8]  : per (c,bin): {in_cw, 1/bw, in_ch, bin_h, delta, d_l,
//                                     d_l+d_r-2*delta, 0}
// ---------------------------------------------------------------------------
__global__ void rq_prep(const float* __restrict__ widths,
                        const float* __restrict__ heights,
                        const float* __restrict__ derivs,
                        float* __restrict__ edges_g,
                        float* __restrict__ params_g) {
  const int c = threadIdx.x;
  if (c >= CCH) return;

  float cwE[KB + 1], chE[KB + 1], dv[KB + 1];

  // widths -> x bin edges
  {
    float v[KB];
    float m = -3.4e38f;
#pragma unroll
    for (int k = 0; k < KB; ++k) { v[k] = widths[c * KB + k]; m = fmaxf(m, v[k]); }
    float s = 0.0f;
#pragma unroll
    for (int k = 0; k < KB; ++k) { v[k] = expf(v[k] - m); s += v[k]; }
    const float a = (1.0f - MIN_BW * (float)KB) / s;
    float cums[KB + 1];
    cums[0] = 0.0f;
    float cum = 0.0f;
#pragma unroll
    for (int k = 0; k < KB; ++k) { cum += v[k] * a + MIN_BW; cums[k + 1] = cum; }
    const float inv = 1.0f / fmaxf(cums[KB], EPSV);
#pragma unroll
    for (int k = 0; k <= KB; ++k) cwE[k] = 2.0f * TAILV * (cums[k] * inv) - TAILV;
  }
  // heights -> y bin edges
  {
    float v[KB];
    float m = -3.4e38f;
#pragma unroll
    for (int k = 0; k < KB; ++k) { v[k] = heights[c * KB + k]; m = fmaxf(m, v[k]); }
    float s = 0.0f;
#pragma unroll
    for (int k = 0; k < KB; ++k) { v[k] = expf(v[k] - m); s += v[k]; }
    const float a = (1.0f - MIN_BH * (float)KB) / s;
    float cums[KB + 1];
    cums[0] = 0.0f;
    float cum = 0.0f;
#pragma unroll
    for (int k = 0; k < KB; ++k) { cum += v[k] * a + MIN_BH; cums[k + 1] = cum; }
    const float inv = 1.0f / fmaxf(cums[KB], EPSV);
#pragma unroll
    for (int k = 0; k <= KB; ++k) chE[k] = 2.0f * TAILV * (cums[k] * inv) - TAILV;
  }
  // derivatives -> softplus + MIN_D  (stable: max(x,0)+log1p(exp(-|x|)))
#pragma unroll
  for (int k = 0; k <= KB; ++k) {
    float x = derivs[c * (KB + 1) + k];
    dv[k] = fmaxf(x, 0.0f) + log1pf(expf(-fabsf(x))) + MIN_D;
  }

#pragma unroll
  for (int k = 0; k < KB; ++k) edges_g[c * KB + k] = cwE[k + 1];

#pragma unroll
  for (int b = 0; b < KB; ++b) {
    const float bw = cwE[b + 1] - cwE[b];
    const float bh = chE[b + 1] - chE[b];
    const float inv_bw = 1.0f / bw;
    const float delta = bh * inv_bw;
    const float dl = dv[b], dr = dv[b + 1];
    float* p = params_g + (c * KB + b) * 8;
    p[0] = cwE[b];
    p[1] = inv_bw;
    p[2] = chE[b];
    p[3] = bh;
    p[4] = delta;
    p[5] = dl;
    p[6] = dl + dr - 2.0f * delta;
    p[7] = 0.0f;
  }
}

// ---------------------------------------------------------------------------
// Spline evaluation of one v4f (4 consecutive channels c0..c0+3).
// ---------------------------------------------------------------------------
__device__ __forceinline__ v4f spline4(const v4f x4, const float (&e)[4][KB],
                                       const float* s_params, int c0) {
  v4f o;
#pragma unroll
  for (int j = 0; j < 4; ++j) {
    const float raw = x4[j];
    const float x = fminf(fmaxf(raw, -TAILV + EPSV), TAILV - EPSV);
    // searchsorted over cw[1..7] -> bin in [0,7], pure VALU (edges in VGPRs)
    int b = 0;
#pragma unroll
    for (int k = 0; k < 7; ++k) b += (x >= e[j][k]) ? 1 : 0;
    const int p = (((c0 + j) << 3) + b) << 3;  // ((c*8)+bin)*8 floats
    const v4f P0 = *(const v4f*)&s_params[p];      // in_cw,1/bw,in_ch,bin_h
    const v4f P1 = *(const v4f*)&s_params[p + 4];  // delta,d_l,coef,0
    const float th = fminf(fmaxf((x - P0[0]) * P0[1], 0.0f), 1.0f);
    const float th2 = th * th;
    const float t1m = th - th2;
    const float num = P0[3] * fmaf(P1[0], th2, P1[1] * t1m);
    const float den = fmaf(P1[2], t1m, P1[0]);
    float r0 = __builtin_amdgcn_rcpf(den);
    r0 = r0 * fmaf(-den, r0, 2.0f);  // 1 Newton step
    const float res = fmaf(num, r0, P0[2]);
    o[j] = (raw > TAILV || raw < -TAILV) ? raw : res;
  }
  return o;
}

// ---------------------------------------------------------------------------
// Apply: HBM-streaming spline evaluation.
//   - params table (32KB) staged to LDS via async global->LDS (ASYNCcnt path)
//   - per-lane fixed channel quad: edges kept in registers -> VALU search
//   - 2x ds_load_b128 parameter gather per element, rcp+Newton division
//   - non-temporal 128-bit stream loads/stores, 2-quad unroll for MLP
// ---------------------------------------------------------------------------
__global__ __launch_bounds__(256) void rq_apply(
    const float* __restrict__ in, float* __restrict__ out,
    const float* __restrict__ edges_g, const float* __restrict__ params_g,
    int nquad) {
  __shared__ __align__(16) float s_params[CCH * KB * 8];  // 32 KB

  const int tid = threadIdx.x;

  // ---- async stage of the parameter table into LDS (CDNA5 async path) ----
  {
    const int nvec = (CCH * KB * 8) / 4;  // 2048 float4, 8 per thread
    for (int i = tid; i < nvec; i += 256) {
      unsigned lds_off =
          (unsigned)(unsigned long long)(lds_float*)(&s_params[i * 4]);
      const float* g = params_g + i * 4;
      asm volatile("global_load_async_to_lds_b128 %0, %1, off"
                   :: "v"(lds_off), "v"(g)
                   : "memory");
    }
    asm volatile("s_wait_asynccnt 0" ::: "memory");
    __syncthreads();
  }

  // ---- per-lane fixed channels: load their 8 edges each into registers ----
  const int c0 = (tid & 31) * 4;  // blockDim*4 % 128 == 0 -> invariant
  float e[4][KB];
#pragma unroll
  for (int r = 0; r < 4; ++r) {
    const v4f* p = (const v4f*)(edges_g + (c0 + r) * KB);
    const v4f a = p[0], b = p[1];
    e[r][0] = a[0]; e[r][1] = a[1]; e[r][2] = a[2]; e[r][3] = a[3];
    e[r][4] = b[0]; e[r][5] = b[1]; e[r][6] = b[2]; e[r][7] = b[3];
  }

  const v4f* in4 = (const v4f*)in;
  v4f* out4 = (v4f*)out;
  const int stride = gridDim.x * blockDim.x;

  int q = blockIdx.x * blockDim.x + tid;
  for (; q < nquad; q += 2 * stride) {
    const int q2 = q + stride;
    const bool have2 = q2 < nquad;
    // issue both stream loads up front (independent b128, NT hint)
    const v4f a = __builtin_nontemporal_load(in4 + q);
    const v4f b = have2 ? __builtin_nontemporal_load(in4 + q2) : a;
    const v4f oa = spline4(a, e, s_params, c0);
    __builtin_nontemporal_store(oa, out4 + q);
    if (have2) {
      const v4f ob = spline4(b, e, s_params, c0);
      __builtin_nontemporal_store(ob, out4 + q2);
    }
  }
}

// ---------------------------------------------------------------------------
extern "C" void kernel_launch(void* const* d_in, const int* in_sizes, int n_in,
                              void* d_out, int out_size, void* d_ws,
                              size_t ws_size, hipStream_t stream) {
  const float* inputs = (const float*)d_in[0];
  const float* widths = (const float*)d_in[1];
  const float* heights = (const float*)d_in[2];
  const float* derivs = (const float*)d_in[3];

  float* edges_g = (float*)d_ws;              // C*K floats   (4 KB)
  float* params_g = edges_g + CCH * KB;       // C*K*8 floats (32 KB)

  rq_prep<<<1, 128, 0, stream>>>(widths, heights, derivs, edges_g, params_g);

  const int nquad = out_size >> 2;  // out_size divisible by 4 (C=128)
  int blocks = (nquad + 255) / 256;
  if (blocks > 2048) blocks = 2048;
  rq_apply<<<blocks, 256, 0, stream>>>(inputs, (float*)d_out, edges_g, params_g,
                                       nquad);
}